// MOERouter_72335839199353
// MI455X (gfx1250) — compile-verified
//
#include <hip/hip_runtime.h>
#include <hip/hip_bf16.h>

// ---------------------------------------------------------------------------
// MoE router for MI455X (gfx1250, wave32).
//
// Roofline: 134 MB of hidden-state reads @ 23.3 TB/s => ~5.8 us floor; the
// 4.3 GFLOP gate GEMM is negligible, so we keep exact f32 semantics with
// V_WMMA_F32_16X16X4_F32. Only 1024 waves exist (16-token tiles), so HBM
// saturation comes from per-wave bytes-in-flight:
//   * staging loads are issued BEFORE the A-chunk loads (FIFO loadcnt),
//   * the workgroup barrier is the raw CDNA5 split barrier preceded only by
//     s_wait_dscnt 0 -- __syncthreads() would drain loadcnt to 0 and kill the
//     16 in-flight A loads; only the LDS stores need to be visible,
//   * each wave keeps a full K=128 chunk of A (16 x global_load_b128) in
//     flight across the barrier, retiring under the WMMA loop, and
//   * the NEXT chunk is warmed into L2 with global_prefetch_b8 during the
//     compute phase (double-buffering effect with zero VGPR cost).
//
// gate_w is staged in LDS row-pair interleaved: (r,c) -> sB[(r/2)*PS+2c+(r&1)]
// so every WMMA B fragment is ONE ds_load_b64 straight into the operand pair
// (compiler fuses pairs into ds_load_2addr_b64). PS=144: 2*PS == 32 (mod 64)
// -> the two wave halves (pairs p and p+2) hit disjoint bank groups.
// ---------------------------------------------------------------------------

typedef float v2f __attribute__((ext_vector_type(2)));
typedef float v8f __attribute__((ext_vector_type(8)));

#define TOKENS 16384
#define DIM    2048
#define NEXP   64
#define KC     128          // K-chunk staged per iteration
#define NPAIR  (KC / 2)     // 64 row pairs
#define PS     144          // dwords per row pair (128 data + 16 pad)

// D = A(16x4) * B(4x16) + C, f32.
// (neg_a, A, neg_b, B, c_mod, C, reuse_a, reuse_b)
static __device__ __forceinline__ v8f wmma_f32_k4(v2f a, v2f b, v8f c) {
    return __builtin_amdgcn_wmma_f32_16x16x4_f32(
        false, a, false, b, (short)0, c, false, false);
}

// Workgroup barrier that releases only LDS traffic (dscnt), leaving global
// loads in flight. __syncthreads() conservatively drains loadcnt too.
static __device__ __forceinline__ void wg_barrier_lds_only() {
    asm volatile("s_wait_dscnt 0x0\n\t"
                 "s_barrier_signal -1\n\t"
                 "s_barrier_wait -1" ::: "memory");
}

// --------------------------- Kernel 1: gate GEMM ---------------------------
__global__ __launch_bounds__(256)
void moe_gate_gemm_wmma(const float* __restrict__ x,
                        const float* __restrict__ gw,
                        const float* __restrict__ gb,
                        float* __restrict__ logits)
{
    __shared__ float sB[NPAIR * PS];             // 36 KB

    const int t    = threadIdx.x;
    const int lane = t & 31;
    const int wave = t >> 5;
    const int tile = blockIdx.x * 8 + wave;      // 16-token tile id
    const int m    = lane & 15;                  // M (A) / N (B,C) sub-index
    const int half = lane >> 4;                  // which half of the wave

    const float* __restrict__ xrow = x + (size_t)(tile * 16 + m) * DIM;

    v8f acc0 = {0.f,0.f,0.f,0.f,0.f,0.f,0.f,0.f};
    v8f acc1 = acc0, acc2 = acc0, acc3 = acc0;

    for (int kc = 0; kc < DIM; kc += KC) {
        // ---- (1) staging loads FIRST (so later waits never drain abuf) ----
        // 64 pairs x 16 col-quads = 1024 items; 4 per thread.
        float4 r0[4], r1[4];
        #pragma unroll
        for (int it = 0; it < 4; ++it) {
            const int w = it * 256 + t;
            const int p = w >> 4;                // row pair
            const int q = w & 15;                // col quad
            const float* __restrict__ g0 =
                gw + (size_t)(kc + 2 * p) * NEXP + 4 * q;
            r0[it] = *(const float4*)g0;
            r1[it] = *(const float4*)(g0 + NEXP);
        }

        // ---- (2) issue the whole A chunk: 16 b128 stay in flight across
        //          the LDS stores AND the barrier (issued after staging). ----
        float4 abuf[16];
        {
            const float4* __restrict__ xr4 = (const float4*)(xrow + kc);
            #pragma unroll
            for (int j = 0; j < 16; ++j)
                abuf[j] = xr4[2 * j + half];
        }

        // ---- (3) interleave row pairs -> contiguous b128 LDS stores -------
        #pragma unroll
        for (int it = 0; it < 4; ++it) {
            const int w = it * 256 + t;
            const int p = w >> 4;
            const int q = w & 15;
            float* d = &sB[p * PS + 8 * q];
            *(float4*)(d)     = make_float4(r0[it].x, r1[it].x, r0[it].y, r1[it].y);
            *(float4*)(d + 4) = make_float4(r0[it].z, r1[it].z, r0[it].w, r1[it].w);
        }
        wg_barrier_lds_only();      // dscnt only -- abuf loads stay in flight

        // ---- (4) warm next chunk into L2 while the matrix pipe works ------
        // Each lane covers its row's next-chunk bytes with two 128B lines.
        {
            const int kcn = (kc + KC) & (DIM - 1);   // wraps on last chunk
            const float* pf = xrow + kcn + half * 64;
            __builtin_prefetch(pf, 0, 1);
            __builtin_prefetch(pf + 32, 0, 1);
        }

        // ---- (5) 128 WMMAs over the chunk; abuf waits retire gradually ----
        #pragma unroll
        for (int j = 0; j < 16; ++j) {           // K = 8j .. 8j+7
            const float4 af = abuf[j];
            const v2f a01 = {af.x, af.y};        // instance A fragment
            const v2f a23 = {af.z, af.w};        // instance B fragment

            // Instance A: K-set {k0,k0+1} (half0) u {k0+4,k0+5} (half1)
            //   -> lane's B rows are exactly pair pA = 4j + 2*half.
            // Instance B: the complementary K-set -> pair pA + 1.
            const int pA = 4 * j + 2 * half;
            const float* __restrict__ bb = &sB[pA * PS + 2 * m];

            v2f b;
            b = *(const v2f*)(bb);                       // eb 0, inst A
            acc0 = wmma_f32_k4(a01, b, acc0);
            b = *(const v2f*)(bb + 32);                  // eb 1
            acc1 = wmma_f32_k4(a01, b, acc1);
            b = *(const v2f*)(bb + 64);                  // eb 2
            acc2 = wmma_f32_k4(a01, b, acc2);
            b = *(const v2f*)(bb + 96);                  // eb 3
            acc3 = wmma_f32_k4(a01, b, acc3);

            b = *(const v2f*)(bb + PS);                  // eb 0, inst B
            acc0 = wmma_f32_k4(a23, b, acc0);
            b = *(const v2f*)(bb + PS + 32);             // eb 1
            acc1 = wmma_f32_k4(a23, b, acc1);
            b = *(const v2f*)(bb + PS + 64);             // eb 2
            acc2 = wmma_f32_k4(a23, b, acc2);
            b = *(const v2f*)(bb + PS + 96);             // eb 3
            acc3 = wmma_f32_k4(a23, b, acc3);
        }
        wg_barrier_lds_only();      // our ds_loads retired before re-staging
    }

    // ---- epilogue: bias + store. C/D layout: VGPR i <-> M = i + 8*half,
    // N = m (+16 per expert block). ----
    const float bias0 = gb[m];
    const float bias1 = gb[m + 16];
    const float bias2 = gb[m + 32];
    const float bias3 = gb[m + 48];
    float* __restrict__ outBase = logits + (size_t)(tile * 16) * NEXP;
    #pragma unroll
    for (int i = 0; i < 8; ++i) {
        const int M = i + 8 * half;
        float* o = outBase + (size_t)M * NEXP + m;
        o[0]  = acc0[i] + bias0;
        o[16] = acc1[i] + bias1;
        o[32] = acc2[i] + bias2;
        o[48] = acc3[i] + bias3;
    }
}

// ---------------------- Kernel 2: softmax + top-2 --------------------------
// Wave32 per token: 2 experts per lane, shfl_xor tree reductions.
__global__ __launch_bounds__(256)
void moe_router_topk(const float* __restrict__ logits,
                     float* __restrict__ norm_out,
                     float* __restrict__ idx_out)
{
    const int lane = threadIdx.x & 31;
    const int wave = threadIdx.x >> 5;
    const int t    = blockIdx.x * 8 + wave;

    const float* __restrict__ l = logits + (size_t)t * NEXP;
    const float l0 = l[lane];
    const float l1 = l[lane + 32];

    // stable softmax: wave max
    float mx = fmaxf(l0, l1);
    #pragma unroll
    for (int s = 16; s > 0; s >>= 1) mx = fmaxf(mx, __shfl_xor(mx, s, 32));

    const float e0 = __expf(l0 - mx);
    const float e1 = __expf(l1 - mx);
    float sum = e0 + e1;
    #pragma unroll
    for (int s = 16; s > 0; s >>= 1) sum += __shfl_xor(sum, s, 32);

    const float p0 = e0 / sum;
    const float p1 = e1 / sum;

    // top-1 (ties -> lower index, matching jax.lax.top_k)
    float v1 = p0; int i1 = lane;
    if (p1 > v1) { v1 = p1; i1 = lane + 32; }
    #pragma unroll
    for (int s = 16; s > 0; s >>= 1) {
        float ov = __shfl_xor(v1, s, 32);
        int   oi = __shfl_xor(i1, s, 32);
        if (ov > v1 || (ov == v1 && oi < i1)) { v1 = ov; i1 = oi; }
    }

    // top-2: exclude i1
    float v2 = -3.0e38f; int i2 = 0x3fffffff;
    if (lane != i1)                    { v2 = p0; i2 = lane; }
    if ((lane + 32) != i1 && p1 > v2)  { v2 = p1; i2 = lane + 32; }
    #pragma unroll
    for (int s = 16; s > 0; s >>= 1) {
        float ov = __shfl_xor(v2, s, 32);
        int   oi = __shfl_xor(i2, s, 32);
        if (ov > v2 || (ov == v2 && oi < i2)) { v2 = ov; i2 = oi; }
    }

    if (lane == 0) {
        const float s2 = fmaxf(v1 + v2, 1e-6f);
        norm_out[(size_t)t * 2 + 0] = v1 / s2;
        norm_out[(size_t)t * 2 + 1] = v2 / s2;
        idx_out [(size_t)t * 2 + 0] = (float)i1;
        idx_out [(size_t)t * 2 + 1] = (float)i2;
    }
}

// ------------------------------- launcher ----------------------------------
extern "C" void kernel_launch(void* const* d_in, const int* in_sizes, int n_in,
                              void* d_out, int out_size, void* d_ws, size_t ws_size,
                              hipStream_t stream)
{
    const float* x  = (const float*)d_in[0];   // [4,4096,2048] f32
    const float* gw = (const float*)d_in[1];   // [2048,64] f32
    const float* gb = (const float*)d_in[2];   // [64] f32

    float* out    = (float*)d_out;
    float* logits = out;                                   // 16384*64
    float* norm   = out + (size_t)TOKENS * NEXP;           // 16384*2
    float* idx    = norm + (size_t)TOKENS * 2;             // 16384*2

    // 1024 token-tiles, 8 waves/block -> 128 blocks
    moe_gate_gemm_wmma<<<TOKENS / 16 / 8, 256, 0, stream>>>(x, gw, gb, logits);
    // wave per token -> 2048 blocks
    moe_router_topk<<<TOKENS / 8, 256, 0, stream>>>(logits, norm, idx);
}